// SO4LR_45157286150747
// MI455X (gfx1250) — compile-verified
//
#include <hip/hip_runtime.h>
#include <math.h>

// SO4LR left TT core: out is the flattened (1,2,2,3) tensor.
//   out[0]  = 1.0f
//   out[10] = cos(theta)
//   out[11] = sin(theta)
//   all other elements = 0.0f
//
// 12 floats total -> single wave32, lane i writes out[i] (i < 12) as one
// coalesced 48B store. theta[0] is a broadcast scalar load. sincosf gives
// fp32-accurate results matching the JAX reference. Every output element is
// written on every call (d_out is poisoned by the harness), and the kernel
// is stateless/deterministic.
__global__ void so4lr_left_core_kernel(const float* __restrict__ theta,
                                       float* __restrict__ out) {
    const int i = (int)threadIdx.x;

    const float t = theta[0];
    float s, c;
    sincosf(t, &s, &c);

    // Branchless per-lane value selection.
    float v = 0.0f;
    v = (i == 0)  ? 1.0f : v;
    v = (i == 10) ? c    : v;
    v = (i == 11) ? s    : v;

    if (i < 12) {
        out[i] = v;
    }
}

extern "C" void kernel_launch(void* const* d_in, const int* in_sizes, int n_in,
                              void* d_out, int out_size, void* d_ws, size_t ws_size,
                              hipStream_t stream) {
    (void)in_sizes; (void)n_in; (void)d_ws; (void)ws_size; (void)out_size;

    const float* theta = (const float*)d_in[0];
    float* out = (float*)d_out;

    // One workgroup of one wave32 covers all 12 outputs.
    so4lr_left_core_kernel<<<dim3(1), dim3(32), 0, stream>>>(theta, out);
}